// AEEncoder_45011257262636
// MI455X (gfx1250) — compile-verified
//
#include <hip/hip_runtime.h>
#include <hip/hip_bf16.h>
#include <math.h>

// Problem constants (match reference)
#define Bn     512
#define IN_F   20000
#define OUT_Fc 1000
#define NNZc   500000
#define CHUNK  4096
#define NCHUNK ((NNZc + CHUNK - 1) / CHUNK)   // 123
#define NGROUP ((OUT_Fc + 15) / 16)           // 63 groups of 16 outputs
#define NNZ_PAD (NNZc + 8 * NGROUP)           // room for per-group padding to x8

typedef __attribute__((ext_vector_type(2))) float v2f;
typedef __attribute__((ext_vector_type(8))) float v8f;

// ---------------------------------------------------------------------------
// K1: transpose x[B][IN_F] -> xT[IN_F][B] so gathered columns are contiguous
// over the batch dimension (coalesced gathers in the WMMA kernel).
// ---------------------------------------------------------------------------
__global__ void k_transpose(const float* __restrict__ x, float* __restrict__ xT) {
    __shared__ float tile[32][33];
    int i0 = blockIdx.x * 32;          // IN_F tile
    int b0 = blockIdx.y * 32;          // batch tile
    int tx = threadIdx.x, ty = threadIdx.y;   // 32 x 8
    #pragma unroll
    for (int r = ty; r < 32; r += 8) {
        int b = b0 + r, i = i0 + tx;
        tile[r][tx] = (i < IN_F) ? x[(size_t)b * IN_F + i] : 0.0f;
    }
    __syncthreads();
    #pragma unroll
    for (int r = ty; r < 32; r += 8) {
        int i = i0 + r, b = b0 + tx;
        if (i < IN_F) xT[(size_t)i * Bn + b] = tile[tx][r];
    }
}

// ---------------------------------------------------------------------------
// K2: per-chunk histogram of out_idx (integer atomics -> deterministic).
// ---------------------------------------------------------------------------
__global__ void k_histo(const int* __restrict__ out_idx, int* __restrict__ hist) {
    __shared__ int h[OUT_Fc];
    int c = blockIdx.x;
    for (int i = threadIdx.x; i < OUT_Fc; i += blockDim.x) h[i] = 0;
    __syncthreads();
    int s = c * CHUNK;
    for (int i = threadIdx.x; i < CHUNK; i += blockDim.x) {
        int e = s + i;
        if (e < NNZc) atomicAdd(&h[out_idx[e]], 1);
    }
    __syncthreads();
    for (int i = threadIdx.x; i < OUT_Fc; i += blockDim.x)
        hist[c * OUT_Fc + i] = h[i];
}

// ---------------------------------------------------------------------------
// K3: stable counting-sort scan with GROUP PADDING.
// Each 16-output group's region in the sorted edge array starts at a position
// such that its length is padded up to a multiple of 8; pad slots are filled
// with zero-weight edges here (in=0, col=0, w=0 -> contribute exactly 0).
// Converts hist[c][o] into global start positions; writes per-group
// [begin, padded_end) into grp[]. Integer-only -> deterministic.
// ---------------------------------------------------------------------------
__global__ void k_scan(int* __restrict__ hist, int* __restrict__ grp,
                       int* __restrict__ packed, float* __restrict__ sw) {
    __shared__ int total[OUT_Fc];
    __shared__ int base[OUT_Fc];
    __shared__ int gbeg[NGROUP], gdend[NGROUP], gend[NGROUP];
    int t = threadIdx.x;
    if (t < OUT_Fc) {
        int run = 0;
        for (int c = 0; c < NCHUNK; c++) {
            int idx = c * OUT_Fc + t;
            int v = hist[idx];
            hist[idx] = run;
            run += v;
        }
        total[t] = run;
    }
    __syncthreads();
    if (t == 0) {
        int cur = 0;
        for (int g = 0; g < NGROUP; g++) {
            gbeg[g] = cur;
            int o1 = (g + 1) * 16; if (o1 > OUT_Fc) o1 = OUT_Fc;
            for (int o = g * 16; o < o1; o++) { base[o] = cur; cur += total[o]; }
            gdend[g] = cur;                 // end of real data
            cur = (cur + 7) & ~7;           // pad group length to multiple of 8
            gend[g] = cur;
        }
    }
    __syncthreads();
    if (t < OUT_Fc) {
        int b = base[t];
        for (int c = 0; c < NCHUNK; c++) hist[c * OUT_Fc + t] += b;
    }
    // zero-fill pad slots (at most 7 per group)
    if (t < NGROUP * 8) {
        int g = t >> 3, i = t & 7;
        int pos = gdend[g] + i;
        if (pos < gend[g]) { packed[pos] = 0; sw[pos] = 0.0f; }
    }
    // publish per-group ranges
    if (t < NGROUP) { grp[2 * t] = gbeg[t]; grp[2 * t + 1] = gend[t]; }
}

// ---------------------------------------------------------------------------
// K4: stable scatter. One wave per chunk; within each 32-edge window the rank
// among same-bucket lanes is computed with a shuffle loop (preserves edge
// order -> stable -> deterministic float sums downstream).
// ---------------------------------------------------------------------------
__global__ void k_scatter(const int* __restrict__ out_idx,
                          const int* __restrict__ in_idx,
                          const float* __restrict__ w,
                          const int* __restrict__ hist,
                          int* __restrict__ packed,
                          float* __restrict__ sw) {
    __shared__ int cnt[OUT_Fc];
    int c = blockIdx.x, lane = threadIdx.x;
    for (int i = lane; i < OUT_Fc; i += 32) cnt[i] = hist[c * OUT_Fc + i];
    int s = c * CHUNK;
    for (int g = 0; g < CHUNK; g += 32) {
        int e = s + g + lane;
        bool valid = (e < NNZc);
        int o = valid ? out_idx[e] : -1;
        int before = 0;
        bool last = true;
        #pragma unroll
        for (int j = 0; j < 32; j++) {
            int oj = __shfl(o, j, 32);
            if (o >= 0 && oj == o) {
                if (j < lane) before++;
                else if (j > lane) last = false;
            }
        }
        if (valid) {
            int pos = cnt[o] + before;      // all ds reads precede ds writes (1 wave, program order)
            int col = o & 15;               // column inside the 16-output WMMA group
            packed[pos] = in_idx[e] | (col << 20);
            sw[pos] = w[e];
        }
        if (valid && last) cnt[o] += before + 1;  // one writer per bucket per window
    }
}

// ---------------------------------------------------------------------------
// K5: WMMA compute. One wave per (16-output group, 32-batch tile).
// Two accumulators (two 16-row batch tiles) share the one-hot B operand and
// the metadata fetch; 8 edges (2 K-steps) per iteration -> 4 WMMAs per
// iteration, zero guards (regions padded to x8).
//
// V_WMMA_F32_16X16X4_F32:  C(16x16 f32) += A(16x4, gathered x) *
//                                          B(4x16, one-hot(col)*weight)
// A layout: lane L, M=L&15; vgpr0 K=(L<16?0:2), vgpr1 K=(L<16?1:3).
// B layout: lane L, N=L&15; vgpr0 row K=(L<16?0:2), vgpr1 row K=(L<16?1:3).
// C layout: vgpr j -> M = j + (L<16?0:8), N = L&15.
// ---------------------------------------------------------------------------
__global__ void k_sparse_wmma(const float* __restrict__ xT,
                              const int* __restrict__ packed,
                              const float* __restrict__ sw,
                              const int* __restrict__ grp,
                              const float* __restrict__ bias,
                              float* __restrict__ out) {
    const int g    = blockIdx.x;           // 16-output group
    const int bt   = blockIdx.y << 5;      // 32-row batch tile start
    const int base = g << 4;
    const int lane = threadIdx.x;
    const int bl   = lane & 15;
    const int kk   = (lane >> 4) << 1;     // 0 for lanes 0-15, 2 for lanes 16-31

    const int start = grp[2 * g];
    const int end   = grp[2 * g + 1];      // end - start is a multiple of 8

    v8f c0 = {0.f,0.f,0.f,0.f,0.f,0.f,0.f,0.f};
    v8f c1 = {0.f,0.f,0.f,0.f,0.f,0.f,0.f,0.f};

    for (int p = start; p < end; p += 8) {
        // K-step 0: edges p..p+3
        int   e  = p + kk;
        int   pk0 = packed[e];      float w0 = sw[e];
        int   pk1 = packed[e + 1];  float w1 = sw[e + 1];
        // K-step 1: edges p+4..p+7
        int   f  = p + 4 + kk;
        int   pk2 = packed[f];      float w2 = sw[f];
        int   pk3 = packed[f + 1];  float w3 = sw[f + 1];

        int in0 = pk0 & 0xFFFFF, col0 = pk0 >> 20;
        int in1 = pk1 & 0xFFFFF, col1 = pk1 >> 20;
        int in2 = pk2 & 0xFFFFF, col2 = pk2 >> 20;
        int in3 = pk3 & 0xFFFFF, col3 = pk3 >> 20;

        v2f a0, a1, a2, a3, b0, b1;
        // batch tile 0 (rows bt..bt+15)
        a0.x = xT[in0 * Bn + bt + bl];
        a0.y = xT[in1 * Bn + bt + bl];
        a2.x = xT[in2 * Bn + bt + bl];
        a2.y = xT[in3 * Bn + bt + bl];
        // batch tile 1 (rows bt+16..bt+31)
        a1.x = xT[in0 * Bn + bt + 16 + bl];
        a1.y = xT[in1 * Bn + bt + 16 + bl];
        a3.x = xT[in2 * Bn + bt + 16 + bl];
        a3.y = xT[in3 * Bn + bt + 16 + bl];
        // shared one-hot B operands
        b0.x = (col0 == bl) ? w0 : 0.0f;
        b0.y = (col1 == bl) ? w1 : 0.0f;
        b1.x = (col2 == bl) ? w2 : 0.0f;
        b1.y = (col3 == bl) ? w3 : 0.0f;

        c0 = __builtin_amdgcn_wmma_f32_16x16x4_f32(false, a0, false, b0, (short)0, c0, false, false);
        c1 = __builtin_amdgcn_wmma_f32_16x16x4_f32(false, a1, false, b0, (short)0, c1, false, false);
        c0 = __builtin_amdgcn_wmma_f32_16x16x4_f32(false, a2, false, b1, (short)0, c0, false, false);
        c1 = __builtin_amdgcn_wmma_f32_16x16x4_f32(false, a3, false, b1, (short)0, c1, false, false);
    }

    const int o   = base + bl;
    const int mhi = (lane >> 4) << 3;      // 0 or 8
    if (o < OUT_Fc) {
        const float bi = bias[o];
        const float kScale = 1.0507009873554805f;
        const float kAlpha = 1.6732632423543772f;
        #pragma unroll
        for (int j = 0; j < 8; j++) {
            float v = c0[j] + bi;
            float s = (v > 0.0f) ? v : kAlpha * expm1f(v);
            out[(size_t)(bt + mhi + j) * OUT_Fc + o] = kScale * s;
        }
        #pragma unroll
        for (int j = 0; j < 8; j++) {
            float v = c1[j] + bi;
            float s = (v > 0.0f) ? v : kAlpha * expm1f(v);
            out[(size_t)(bt + 16 + mhi + j) * OUT_Fc + o] = kScale * s;
        }
    }
}

// ---------------------------------------------------------------------------
// Launch
// ---------------------------------------------------------------------------
extern "C" void kernel_launch(void* const* d_in, const int* in_sizes, int n_in,
                              void* d_out, int out_size, void* d_ws, size_t ws_size,
                              hipStream_t stream) {
    const float* x       = (const float*)d_in[0];   // [B, IN_F]
    const float* w       = (const float*)d_in[1];   // [NNZ]
    const float* bias    = (const float*)d_in[2];   // [OUT_F]
    const int*   out_idx = (const int*)d_in[3];     // [NNZ]
    const int*   in_idx  = (const int*)d_in[4];     // [NNZ]
    float*       out     = (float*)d_out;           // [B, OUT_F]

    // Workspace carve (~45.5 MB total)
    float* xT      = (float*)d_ws;                          // IN_F*B floats
    int*   packed  = (int*)(xT + (size_t)IN_F * Bn);        // NNZ_PAD ints
    float* sw      = (float*)(packed + NNZ_PAD);            // NNZ_PAD floats
    int*   grp     = (int*)(sw + NNZ_PAD);                  // 2*NGROUP ints
    int*   hist    = grp + 2 * NGROUP;                      // NCHUNK*OUT_F ints

    k_transpose<<<dim3((IN_F + 31) / 32, Bn / 32), dim3(32, 8), 0, stream>>>(x, xT);
    k_histo<<<NCHUNK, 256, 0, stream>>>(out_idx, hist);
    k_scan<<<1, 1024, 0, stream>>>(hist, grp, packed, sw);
    k_scatter<<<NCHUNK, 32, 0, stream>>>(out_idx, in_idx, w, hist, packed, sw);
    k_sparse_wmma<<<dim3(NGROUP, Bn / 32), 32, 0, stream>>>(
        xT, packed, sw, grp, bias, out);
}